// CRQVAE_30039001268972
// MI455X (gfx1250) — compile-verified
//
#include <hip/hip_runtime.h>
#include <hip/hip_bf16.h>

typedef __attribute__((ext_vector_type(2))) float v2f;
typedef __attribute__((ext_vector_type(8))) float v8f;

namespace {
constexpr int   kN    = 131072;
constexpr int   kD0   = 768;
constexpr int   kD1   = 512;
constexpr int   kD2   = 256;
constexpr int   kD3   = 64;
constexpr int   kCode = 256;
constexpr int   kLvl  = 4;
constexpr float kBeta = 0.25f;

constexpr int kRows  = 32;           // rows per workgroup tile
constexpr int kBlock = 256;          // 8 wave32
constexpr int kGrid  = kN / kRows;   // 4096

// ---- LDS layout (float indices), two overlaid regions ----
constexpr int XS_ST   = kD0 + 4;                 // 772 (pad vs 64-bank conflicts)
constexpr int XS_OFF  = 0;
constexpr int REGION2 = kRows * XS_ST;           // 24704
constexpr int H1_ST   = kD1 + 4;                 // 516
constexpr int H1_OFF  = REGION2;
// region 1 reuse (x tile dead after layer 1):
constexpr int H2_ST   = kD2 + 4;                 // 260
constexpr int H2_OFF  = 0;
constexpr int ZS_ST   = kD3 + 4;                 // 68
constexpr int ZS_OFF  = kRows * H2_ST;           // 8320
constexpr int XQ_OFF  = ZS_OFF + kRows * ZS_ST;  // 10496
// region 2 reuse (h1 dead after layer 2):
constexpr int CN_OFF  = H1_OFF;                  // 4*256 codebook norms
constexpr int SC_ST   = kCode + 4;               // 260
constexpr int SC_OFF  = CN_OFF + kLvl * kCode;
constexpr int RV_OFF  = SC_OFF + kRows * SC_ST;
constexpr int RI_OFF  = RV_OFF + kRows * 8;
constexpr int IX_OFF  = RI_OFF + kRows * 8;
constexpr int SMEM_FLOATS = H1_OFF + kRows * H1_ST;   // 41216 -> 164,864 B
static_assert(IX_OFF + kRows <= SMEM_FLOATS, "LDS overflow");
static_assert(XQ_OFF + kRows * ZS_ST <= REGION2, "region1 overflow");

// ---- output layout: x_q [N,64] f32, rq_loss scalar, codes [L,N] (as float) ----
constexpr long OUT_LOSS  = (long)kN * kD3;
constexpr long OUT_CODES = OUT_LOSS + 1;
} // namespace

__device__ __forceinline__ v8f wmma4f32(v2f a, v2f b, v8f c) {
  // D = A(16x4, f32) * B(4x16, f32) + C(16x16, f32)
  return __builtin_amdgcn_wmma_f32_16x16x4_f32(false, a, false, b, (short)0, c,
                                               false, false);
}

__global__ __launch_bounds__(kBlock) void crqvae_fused(
    const float* __restrict__ x,  const float* __restrict__ W0, const float* __restrict__ b0,
    const float* __restrict__ W1, const float* __restrict__ b1,
    const float* __restrict__ W2, const float* __restrict__ b2,
    const float* __restrict__ cb, float* __restrict__ out,
    float* __restrict__ block_loss) {
  extern __shared__ float smem[];
  const int tid  = threadIdx.x;
  const int lane = tid & 31;
  const int wave = tid >> 5;
  const int m    = lane & 15;    // A row / B col within 16-tile
  const int g    = lane >> 4;    // lane half -> K pair select / M+8 select
  const int rw   = wave >> 2;    // row sub-tile 0..1
  const int cw   = wave & 3;     // col slice  0..3
  const int row0 = rw * 16;
  const long gRow0 = (long)blockIdx.x * kRows;

  // ---- stage x tile with CDNA5 async copy: global -> LDS, no VGPR return ----
  // 32 rows x 768 f32 = 6144 chunks of 16 B. Global row stride 3072 B and LDS
  // row stride 3088 B are both 16-B aligned. No static __shared__ in this
  // kernel, so dynamic-LDS byte offsets start at 0. Tracked by ASYNCcnt.
  {
    for (int c = tid; c < kRows * (kD0 / 4); c += kBlock) {
      const int row   = c / (kD0 / 4);
      const int col16 = c - row * (kD0 / 4);
      const float* src = x + (gRow0 + row) * kD0 + col16 * 4;
      const unsigned lds_off =
          (unsigned)((XS_OFF + row * XS_ST + col16 * 4) * sizeof(float));
      asm volatile("global_load_async_to_lds_b128 %0, %1, off"
                   :: "v"(lds_off), "v"(src) : "memory");
    }
    asm volatile("s_wait_asynccnt 0x0" ::: "memory");
  }
  __syncthreads();

  // ---- layer 1: 768 -> 512, bias + ReLU ; each wave: 16 rows x 128 cols ----
  {
    v8f acc[8];
#pragma unroll
    for (int t = 0; t < 8; ++t) acc[t] = (v8f){0.f,0.f,0.f,0.f,0.f,0.f,0.f,0.f};
    const float* A = smem + XS_OFF + (row0 + m) * XS_ST;
    const int colb = cw * 128;
#pragma unroll 2
    for (int kk = 0; kk < kD0; kk += 4) {
      v2f a; a.x = A[kk + 2*g]; a.y = A[kk + 2*g + 1];
#pragma unroll
      for (int t = 0; t < 8; ++t) {
        const int col = colb + t * 16 + m;
        v2f b; b.x = W0[(kk + 2*g) * kD1 + col]; b.y = W0[(kk + 2*g + 1) * kD1 + col];
        acc[t] = wmma4f32(a, b, acc[t]);
      }
    }
#pragma unroll
    for (int t = 0; t < 8; ++t) {
      const int col = colb + t * 16 + m;
      const float bias = b0[col];
#pragma unroll
      for (int v = 0; v < 8; ++v) {
        const float val = acc[t][v] + bias;
        smem[H1_OFF + (row0 + v + 8*g) * H1_ST + col] = val > 0.f ? val : 0.f;
      }
    }
  }
  __syncthreads();

  // ---- layer 2: 512 -> 256, bias + ReLU ; each wave: 16 rows x 64 cols ----
  {
    v8f acc[4];
#pragma unroll
    for (int t = 0; t < 4; ++t) acc[t] = (v8f){0.f,0.f,0.f,0.f,0.f,0.f,0.f,0.f};
    const float* A = smem + H1_OFF + (row0 + m) * H1_ST;
    const int colb = cw * 64;
#pragma unroll 2
    for (int kk = 0; kk < kD1; kk += 4) {
      v2f a; a.x = A[kk + 2*g]; a.y = A[kk + 2*g + 1];
#pragma unroll
      for (int t = 0; t < 4; ++t) {
        const int col = colb + t * 16 + m;
        v2f b; b.x = W1[(kk + 2*g) * kD2 + col]; b.y = W1[(kk + 2*g + 1) * kD2 + col];
        acc[t] = wmma4f32(a, b, acc[t]);
      }
    }
#pragma unroll
    for (int t = 0; t < 4; ++t) {
      const int col = colb + t * 16 + m;
      const float bias = b1[col];
#pragma unroll
      for (int v = 0; v < 8; ++v) {
        const float val = acc[t][v] + bias;
        smem[H2_OFF + (row0 + v + 8*g) * H2_ST + col] = val > 0.f ? val : 0.f;
      }
    }
  }
  __syncthreads();

  // ---- layer 3: 256 -> 64, bias only ; each wave: 16 rows x 16 cols ----
  {
    v8f acc = (v8f){0.f,0.f,0.f,0.f,0.f,0.f,0.f,0.f};
    const float* A = smem + H2_OFF + (row0 + m) * H2_ST;
    const int colb = cw * 16;
#pragma unroll 2
    for (int kk = 0; kk < kD2; kk += 4) {
      v2f a; a.x = A[kk + 2*g]; a.y = A[kk + 2*g + 1];
      const int col = colb + m;
      v2f b; b.x = W2[(kk + 2*g) * kD3 + col]; b.y = W2[(kk + 2*g + 1) * kD3 + col];
      acc = wmma4f32(a, b, acc);
    }
    const int col = colb + m;
    const float bias = b2[col];
#pragma unroll
    for (int v = 0; v < 8; ++v) {
      const int r = row0 + v + 8*g;
      smem[ZS_OFF + r * ZS_ST + col] = acc[v] + bias;  // residual = z
      smem[XQ_OFF + r * ZS_ST + col] = 0.f;            // x_q accumulator
    }
  }
  __syncthreads();

  // ---- codebook squared norms (4 levels x 256 codes), into region 2 ----
  for (int n = tid; n < kLvl * kCode; n += kBlock) {
    const float* c = cb + n * kD3;
    float s = 0.f;
#pragma unroll 1
    for (int j = 0; j < kD3; ++j) s += c[j] * c[j];
    smem[CN_OFF + n] = s;
  }
  __syncthreads();

  float lsum = 0.f;   // per-thread loss partial, Σ(q - r)^2

  int* const ri  = (int*)(smem + RI_OFF);
  int* const ixp = (int*)(smem + IX_OFF);

  for (int l = 0; l < kLvl; ++l) {
    // ---- distance GEMM: scores = ||c||^2 - 2 * r.cT  (argmin-equivalent) ----
    {
      v8f acc[4];
#pragma unroll
      for (int t = 0; t < 4; ++t) acc[t] = (v8f){0.f,0.f,0.f,0.f,0.f,0.f,0.f,0.f};
      const float* A  = smem + ZS_OFF + (row0 + m) * ZS_ST;
      const float* CB = cb + l * kCode * kD3;
      const int colb = cw * 64;
#pragma unroll 2
      for (int kk = 0; kk < kD3; kk += 4) {
        v2f a; a.x = A[kk + 2*g]; a.y = A[kk + 2*g + 1];
#pragma unroll
        for (int t = 0; t < 4; ++t) {
          const int col = colb + t * 16 + m;           // code index
          v2f b; b.x = CB[col * kD3 + kk + 2*g]; b.y = CB[col * kD3 + kk + 2*g + 1];
          acc[t] = wmma4f32(a, b, acc[t]);
        }
      }
#pragma unroll
      for (int t = 0; t < 4; ++t) {
        const int col = colb + t * 16 + m;
        const float cn = smem[CN_OFF + l * kCode + col];
#pragma unroll
        for (int v = 0; v < 8; ++v)
          smem[SC_OFF + (row0 + v + 8*g) * SC_ST + col] = cn - 2.f * acc[t][v];
      }
    }
    __syncthreads();

    // ---- argmin over 256 codes: 8-thread team per row ----
    {
      const int row = tid >> 3, part = tid & 7;
      float best = 3.402823466e+38f;
      int bi = 0;
      const float* s = smem + SC_OFF + row * SC_ST + part * 32;
#pragma unroll 1
      for (int j = 0; j < 32; ++j) {
        const float v = s[j];
        if (v < best) { best = v; bi = part * 32 + j; }
      }
      smem[RV_OFF + row * 8 + part] = best;
      ri[row * 8 + part] = bi;
    }
    __syncthreads();
    {
      const int row = tid >> 3, part = tid & 7;
      if (part == 0) {
        float best = 3.402823466e+38f;
        int bi = 0;
#pragma unroll
        for (int p = 0; p < 8; ++p) {           // ascending => first-min on ties
          const float v = smem[RV_OFF + row * 8 + p];
          if (v < best) { best = v; bi = ri[row * 8 + p]; }
        }
        ixp[row] = bi;
        out[OUT_CODES + (long)l * kN + gRow0 + row] = (float)bi;  // codes[l][n]
      }
    }
    __syncthreads();

    // ---- gather q, loss, residual update, x_q accumulate ----
    {
      const int row = tid >> 3, jb = (tid & 7) * 8;
      const int idx = ixp[row];
      const float* q = cb + (l * kCode + idx) * kD3 + jb;
#pragma unroll
      for (int jj = 0; jj < 8; ++jj) {
        const float qv = q[jj];
        const float r  = smem[ZS_OFF + row * ZS_ST + jb + jj];
        const float d  = qv - r;
        lsum += d * d;
        smem[ZS_OFF + row * ZS_ST + jb + jj] = r - qv;
        smem[XQ_OFF + row * ZS_ST + jb + jj] += qv;
      }
    }
    __syncthreads();
  }

  // ---- write x_q (contiguous: thread tid covers 8 consecutive floats) ----
  {
    const int row = tid >> 3, jb = (tid & 7) * 8;
    float* dst = out + (gRow0 + row) * kD3 + jb;
#pragma unroll
    for (int jj = 0; jj < 8; ++jj) dst[jj] = smem[XQ_OFF + row * ZS_ST + jb + jj];
  }

  // ---- deterministic per-block loss partial ----
  __syncthreads();
  smem[SC_OFF + tid] = lsum;
  __syncthreads();
  if (tid == 0) {
    float t = 0.f;
    for (int i = 0; i < kBlock; ++i) t += smem[SC_OFF + i];
    block_loss[blockIdx.x] = t;
  }
}

__global__ __launch_bounds__(256) void crqvae_loss_reduce(
    const float* __restrict__ block_loss, float* __restrict__ out) {
  __shared__ float red[256];
  float s = 0.f;
  for (int i = threadIdx.x; i < kGrid; i += 256) s += block_loss[i];
  red[threadIdx.x] = s;
  __syncthreads();
  if (threadIdx.x == 0) {
    float t = 0.f;
    for (int i = 0; i < 256; ++i) t += red[i];
    // rq_loss = (1+beta)/L * Σ_l mean((q-r)^2); mean over N*64 elements
    out[OUT_LOSS] = t * ((1.0f + kBeta) / ((float)kLvl * (float)kN * (float)kD3));
  }
}

extern "C" void kernel_launch(void* const* d_in, const int* in_sizes, int n_in,
                              void* d_out, int out_size, void* d_ws, size_t ws_size,
                              hipStream_t stream) {
  (void)in_sizes; (void)n_in; (void)out_size; (void)ws_size;
  const float* x  = (const float*)d_in[0];
  const float* W0 = (const float*)d_in[1];
  const float* b0 = (const float*)d_in[2];
  const float* W1 = (const float*)d_in[3];
  const float* b1 = (const float*)d_in[4];
  const float* W2 = (const float*)d_in[5];
  const float* b2 = (const float*)d_in[6];
  const float* cb = (const float*)d_in[7];
  float* out = (float*)d_out;
  float* block_loss = (float*)d_ws;   // kGrid floats

  const size_t smem_bytes = (size_t)SMEM_FLOATS * sizeof(float);
  crqvae_fused<<<kGrid, kBlock, smem_bytes, stream>>>(x, W0, b0, W1, b1, W2, b2,
                                                      cb, out, block_loss);
  crqvae_loss_reduce<<<1, 256, 0, stream>>>(block_loss, out);
}